// BaseGrid_19997367730399
// MI455X (gfx1250) — compile-verified
//
#include <hip/hip_runtime.h>

// ---------------------------------------------------------------------------
// NeRF-style dense raymarch + stable compaction for MI455X (gfx1250, wave32).
//
// Pipeline (all bandwidth-bound; recompute ray math instead of staging 400MB):
//   k_count  : per-ray valid-sample count (float, exact)            [1024x256]
//   k_scan1  : per-256-ray exclusive scan via V_WMMA_F32_16X16X4_F32 [1024x32]
//   k_scan2  : scan of 1024 block totals (4 WMMA chunks, 1 wave)     [1x32]
//   k_scatter: recompute + scatter 7 outputs/sample with NT stores   [1024x256]
//
// Scan-by-matmul: inclusive_scan(x[0..255]) with X as 16x16 row-major:
//   rowInc = X * U        (U upper-triangular ones, K sliced into 4 WMMAs)
//   out    = rowInc + Lstrict * broadcast(rowTotals)   (4 more WMMAs)
// f32 WMMA is exact for integer counts < 2^24, so this equals an int scan.
// ---------------------------------------------------------------------------

#define NRAYS 262144
#define S 64
#define M_TOTAL 16777216ULL   // NRAYS * S
#define NBLK1 1024            // NRAYS / 256

typedef __attribute__((ext_vector_type(2))) float v2f;
typedef __attribute__((ext_vector_type(8))) float v8f;

__device__ __forceinline__ float readlane_f(float v, int lane) {
  return __builtin_bit_cast(float, __builtin_amdgcn_readlane(__builtin_bit_cast(int, v), lane));
}

// Inclusive scan of 256 consecutive floats, one wave32, returns C/D-layout v8f:
// lane l, VGPR v holds inclusive[16*(v + 8*(l>>4)) + (l&15)].
__device__ __forceinline__ v8f wmma_scan256(const float* __restrict__ src) {
  const int l = threadIdx.x & 31;
  const int n = l & 15;   // A-row M / B-col N
  const int h = l >> 4;   // lane half selects K pair
  v8f c = {};
  // rowInc = X * U  (K=16 as 4 slices of K=4)
#pragma unroll
  for (int s = 0; s < 4; ++s) {
    v2f a, b;
#pragma unroll
    for (int t = 0; t < 2; ++t) {
      const int kg = 4 * s + 2 * h + t;         // global K index of this slice
      a[t] = src[16 * n + kg];                  // A[M=n][K=kg] = x[16n+kg]
      b[t] = (kg <= n) ? 1.0f : 0.0f;           // U[kg][N=n] (incl. diagonal)
    }
    c = __builtin_amdgcn_wmma_f32_16x16x4_f32(false, a, false, b, (short)0, c,
                                              false, false);
  }
  // Broadcast the 16 row totals (rowInc[r][15] lives in lane 15 / lane 31).
  float rt[16];
#pragma unroll
  for (int q = 0; q < 8; ++q) {
    rt[q]     = readlane_f(c[q], 15);
    rt[q + 8] = readlane_f(c[q], 31);
  }
  // out = Lstrict * Y + rowInc, Y[q][*] = rowTotal[q]
#pragma unroll
  for (int s = 0; s < 4; ++s) {
    v2f a, b;
#pragma unroll
    for (int t = 0; t < 2; ++t) {
      const int kg = 4 * s + 2 * h + t;
      a[t] = (kg < n) ? 1.0f : 0.0f;                     // Lstrict[M=n][kg]
      b[t] = h ? rt[4 * s + 2 + t] : rt[4 * s + t];      // rowTotal[kg]
    }
    c = __builtin_amdgcn_wmma_f32_16x16x4_f32(false, a, false, b, (short)0, c,
                                              false, false);
  }
  return c;
}

// Reference ray/AABB math (matches jnp ops exactly).
__device__ __forceinline__ void ray_setup(const float* __restrict__ ro,
                                          const float* __restrict__ rd, int r,
                                          float radius, float& ox, float& oy,
                                          float& oz, float& dx, float& dy,
                                          float& dz, float& nr, float& fr) {
  ox = ro[3 * r + 0]; oy = ro[3 * r + 1]; oz = ro[3 * r + 2];
  dx = rd[3 * r + 0]; dy = rd[3 * r + 1]; dz = rd[3 * r + 2];
  float n0, f0;
  {
    const float dd = (dx == 0.0f) ? 1e-6f : dx;
    const float inv = 1.0f / dd;
    const float t0 = (-radius - ox) * inv, t1 = (radius - ox) * inv;
    n0 = fminf(t0, t1); f0 = fmaxf(t0, t1);
  }
  {
    const float dd = (dy == 0.0f) ? 1e-6f : dy;
    const float inv = 1.0f / dd;
    const float t0 = (-radius - oy) * inv, t1 = (radius - oy) * inv;
    n0 = fmaxf(n0, fminf(t0, t1)); f0 = fminf(f0, fmaxf(t0, t1));
  }
  {
    const float dd = (dz == 0.0f) ? 1e-6f : dz;
    const float inv = 1.0f / dd;
    const float t0 = (-radius - oz) * inv, t1 = (radius - oz) * inv;
    n0 = fmaxf(n0, fminf(t0, t1)); f0 = fminf(f0, fmaxf(t0, t1));
  }
  nr = n0; fr = f0;
}

__global__ void __launch_bounds__(256) k_count(const float* __restrict__ ro,
                                               const float* __restrict__ rd,
                                               const float* __restrict__ radii,
                                               float* __restrict__ cnt) {
  const int r = blockIdx.x * 256 + threadIdx.x;
  if (r >= NRAYS) return;
  if (r + 8192 < NRAYS) {  // stream-ahead prefetch -> global_prefetch_b8
    __builtin_prefetch(ro + 3 * (r + 8192), 0, 1);
    __builtin_prefetch(rd + 3 * (r + 8192), 0, 1);
  }
  const float radius = radii[0];
  float ox, oy, oz, dx, dy, dz, nr, fr;
  ray_setup(ro, rd, r, radius, ox, oy, oz, dx, dy, dz, nr, fr);
  const float invR = 1.0f / radius;
  int cv = 0;
#pragma unroll 8
  for (int j = 0; j < S; ++j) {
    const float t = (float)j * (1.0f / 63.0f);
    const float dep = nr + (fr - nr) * t;
    const float sx = (ox + dx * dep) * invR;
    const float sy = (oy + dy * dep) * invR;
    const float sz = (oz + dz * dep) * invR;
    cv += (sx * sx + sy * sy + sz * sz <= 1.0f) ? 1 : 0;
  }
  cnt[r] = (float)cv;
}

__global__ void __launch_bounds__(32) k_scan1(const float* __restrict__ cnt,
                                              float* __restrict__ rayExcl,
                                              float* __restrict__ blockTot) {
  const int base = blockIdx.x * 256;
  const v8f c = wmma_scan256(cnt + base);
  const int l = threadIdx.x & 31, n = l & 15, h = l >> 4;
#pragma unroll
  for (int v = 0; v < 8; ++v) {
    const int idx = 16 * (v + 8 * h) + n;
    rayExcl[base + idx] = c[v] - cnt[base + idx];  // exclusive = incl - self
  }
  if (l == 31) blockTot[blockIdx.x] = c[7];  // inclusive[255]
}

__global__ void __launch_bounds__(32) k_scan2(const float* __restrict__ blockTot,
                                              float* __restrict__ blockExcl,
                                              float* __restrict__ nvOut,
                                              float* __restrict__ out) {
  const int l = threadIdx.x & 31, n = l & 15, h = l >> 4;
  float carry = 0.0f;
  for (int ch = 0; ch < 4; ++ch) {  // 1024 totals = 4 x 256, serial carry
    const int base = ch * 256;
    const v8f c = wmma_scan256(blockTot + base);
#pragma unroll
    for (int v = 0; v < 8; ++v) {
      const int idx = 16 * (v + 8 * h) + n;
      blockExcl[base + idx] = carry + c[v] - blockTot[base + idx];
    }
    carry += readlane_f(c[7], 31);
  }
  if (l == 0) {
    nvOut[0] = carry;                    // num_valid for the scatter pass
    out[7ULL * M_TOTAL] = carry;         // num_valid output (last element)
  }
}

__global__ void __launch_bounds__(256) k_scatter(
    const float* __restrict__ ro, const float* __restrict__ rd,
    const float* __restrict__ radii, const float* __restrict__ rayExcl,
    const float* __restrict__ blockExcl, const float* __restrict__ nvP,
    float* __restrict__ out) {
  const int r = blockIdx.x * 256 + threadIdx.x;
  if (r >= NRAYS) return;
  if (r + 8192 < NRAYS) {
    __builtin_prefetch(ro + 3 * (r + 8192), 0, 1);
    __builtin_prefetch(rd + 3 * (r + 8192), 0, 1);
  }
  const float radius = radii[0];
  float ox, oy, oz, dx, dy, dz, nr, fr;
  ray_setup(ro, rd, r, radius, ox, oy, oz, dx, dy, dz, nr, fr);
  const float invR = 1.0f / radius;

  const long base = (long)(rayExcl[r] + blockExcl[r >> 8]);  // valid before ray r
  const long numValid = (long)nvP[0];
  const size_t M = (size_t)M_TOTAL;
  const float fr_idx = (float)r;

  int lv = 0;                 // valid samples emitted so far within this ray
  float prevDep = nr;         // diff(depth, prepend=near) -> delta[0] == 0
#pragma unroll 4
  for (int j = 0; j < S; ++j) {
    const float t = (float)j * (1.0f / 63.0f);
    const float dep = nr + (fr - nr) * t;
    const float sx = (ox + dx * dep) * invR;
    const float sy = (oy + dy * dep) * invR;
    const float sz = (oz + dz * dep) * invR;
    const bool m = (sx * sx + sy * sy + sz * sz) <= 1.0f;
    const float delta = dep - prevDep;
    prevDep = dep;

    long k;
    float bnd;
    if (m) {
      k = base + lv;
      ++lv;
      bnd = (lv == 1) ? 1.0f : 0.0f;  // pack boundary: first valid of the ray
    } else {
      // stable invalid tail: num_valid + (flat_idx - valid_before(flat_idx))
      k = numValid + (long)(r * S + j) - (base + (long)lv);
      bnd = 0.0f;
    }
    const size_t ks = (size_t)k;
    // Outputs never re-read & exceed 192MB L2 -> non-temporal stores.
    __builtin_nontemporal_store(fr_idx, out + ks);                    // ridx
    __builtin_nontemporal_store(sx, out + M + 3 * ks + 0);            // samples
    __builtin_nontemporal_store(sy, out + M + 3 * ks + 1);
    __builtin_nontemporal_store(sz, out + M + 3 * ks + 2);
    __builtin_nontemporal_store(dep, out + 4 * M + ks);               // depth
    __builtin_nontemporal_store(delta, out + 5 * M + ks);             // deltas
    __builtin_nontemporal_store(bnd, out + 6 * M + ks);               // boundary
  }
}

extern "C" void kernel_launch(void* const* d_in, const int* in_sizes, int n_in,
                              void* d_out, int out_size, void* d_ws,
                              size_t ws_size, hipStream_t stream) {
  (void)in_sizes; (void)n_in; (void)out_size; (void)ws_size;
  const float* ro = (const float*)d_in[0];
  const float* rd = (const float*)d_in[1];
  const float* radii = (const float*)d_in[2];
  float* out = (float*)d_out;

  float* ws = (float*)d_ws;          // ~2.1 MB scratch, all floats (exact ints)
  float* cnt = ws;                   // [NRAYS]  per-ray valid count
  float* rayExcl = cnt + NRAYS;      // [NRAYS]  excl. prefix within 256-block
  float* blockTot = rayExcl + NRAYS; // [NBLK1]  per-block totals
  float* blockExcl = blockTot + NBLK1; // [NBLK1] excl. prefix of block totals
  float* nv = blockExcl + NBLK1;     // [1]      num_valid

  k_count<<<NRAYS / 256, 256, 0, stream>>>(ro, rd, radii, cnt);
  k_scan1<<<NBLK1, 32, 0, stream>>>(cnt, rayExcl, blockTot);
  k_scan2<<<1, 32, 0, stream>>>(blockTot, blockExcl, nv, out);
  k_scatter<<<NRAYS / 256, 256, 0, stream>>>(ro, rd, radii, rayExcl, blockExcl,
                                             nv, out);
}